// IncrementalRBFLayer_90005334655649
// MI455X (gfx1250) — compile-verified
//
#include <hip/hip_runtime.h>
#include <math.h>

// ---------------------------------------------------------------------------
// out[b,o] = exp(-0.5 * (x_b - m_:,o)^T * (S_o .^ 2) * (x_b - m_:,o))
//
// q[b,o] = sum_ij A_ij x_i x_j - x_b.v_o + c_o,  A_o = S_o.^2 (elementwise)
// Symmetric-shift decomposition (C = A + A^T, shift_d == shift_{64-d}):
//   sum_ij A_ij x_i x_j = sum_{d=0}^{33} sum_i Bsym[d,i] * x_i * x_{(i+d)%64}
//     Bsym[0,i]  = S_ii^2
//     Bsym[d,i]  = S[i][j]^2 + S[j][i]^2          (d=1..31, j=(i+d)%64)
//     Bsym[32,i] = 0.5*(S[i][j]^2 + S[j][i]^2)
// => f32 GEMM with K = 34*64 + 64 + 4 via V_WMMA_F32_16X16X4_F32:
//    Paug row b: [ x_i*x_{(i+d)%64} (2176) | x_k (64) | 1 0 0 0 ]
//    Baug col o: [ Bsym               (2176) | -v_o   | c0 0 0 0 ]
//   v_o = (A_o + A_o^T) m_o ; c_o = m_o^T A_o m_o
// Baug prebuilt in WMMA-fragment order in d_ws (18.4 MB, L2-resident);
// Paug synthesized on the fly from LDS-staged (doubled) x rows.
// ---------------------------------------------------------------------------

typedef __attribute__((ext_vector_type(2))) float v2f;
typedef __attribute__((ext_vector_type(8))) float v8f;

#define B_DIM    2048
#define IN_DIM   64
#define OUT_DIM  2048
#define ND       34                      // shifts d = 0..33
#define KQ_MAIN  (ND * 16)               // 544 k-quads for the symmetric part
#define KQ_TOT   (KQ_MAIN + 16 + 1)      // + 16 quads of -v + 1 quad (c0,0,0,0)
#define OT       (OUT_DIM / 16)          // 128 o-tiles
#define XS_STRIDE 132                    // doubled row (128) + pad, conflict-free

// ---------------------------------------------------------------------------
// Precompute Baug in fragment-major order:
// Bf[(ot*KQ_TOT + q)*32 + lane] = float2 of
//   Baug[o = ot*16 + (lane&15)][k = 4q + 2*(lane>>4) + {0,1}]
// ---------------------------------------------------------------------------
__global__ __launch_bounds__(256) void build_bfrag(
    const float* __restrict__ S,      // inv_std_dev [OUT][64][64]
    const float* __restrict__ means,  // [64][OUT]
    float2* __restrict__ Bf) {
  const long total = (long)OT * KQ_TOT * 32;
  long flat = (long)blockIdx.x * blockDim.x + threadIdx.x;
  if (flat >= total) return;

  const int lane = (int)(flat & 31);
  const long t   = flat >> 5;
  const int q    = (int)(t % KQ_TOT);
  const int ot   = (int)(t / KQ_TOT);
  const int hi   = lane >> 4;
  const int o    = ot * 16 + (lane & 15);
  const float* Ao = S + ((long)o << 12);   // S_o, 64x64 row-major

  float2 r;
  if (q < KQ_MAIN) {
    const int d  = q >> 4;
    const int i0 = ((q & 15) << 2) + 2 * hi;   // i for .x ; i+1 for .y
    if (d == 0) {
      const float s0 = Ao[(i0 << 6) + i0];
      const float s1 = Ao[((i0 + 1) << 6) + (i0 + 1)];
      r.x = s0 * s0;
      r.y = s1 * s1;
    } else {
      const float w = (d == 32) ? 0.5f : 1.0f;
      const int j0 = (i0 + d) & 63;
      const int j1 = (i0 + 1 + d) & 63;
      const float a0 = Ao[(i0 << 6) + j0];
      const float b0 = Ao[(j0 << 6) + i0];
      const float a1 = Ao[((i0 + 1) << 6) + j1];
      const float b1 = Ao[(j1 << 6) + (i0 + 1)];
      r.x = w * (a0 * a0 + b0 * b0);
      r.y = w * (a1 * a1 + b1 * b1);
    }
  } else if (q < KQ_MAIN + 16) {
    // -v_o[jj],  v = (A + A^T) m
    const int jj = ((q - KQ_MAIN) << 2) + 2 * hi;
    float v0 = 0.f, v1 = 0.f;
    for (int j = 0; j < 64; ++j) {
      const float mj = means[j * OUT_DIM + o];
      float a;
      a = Ao[(jj << 6) + j];       v0 += a * a * mj;
      a = Ao[(j << 6) + jj];       v0 += a * a * mj;
      a = Ao[((jj + 1) << 6) + j]; v1 += a * a * mj;
      a = Ao[(j << 6) + jj + 1];   v1 += a * a * mj;
    }
    r.x = -v0;
    r.y = -v1;
  } else {
    // (c0, 0) for lanes 0-15, (0, 0) for lanes 16-31
    if (hi) {
      r.x = 0.f; r.y = 0.f;
    } else {
      float c0 = 0.f;
      for (int i = 0; i < 64; ++i) {
        const float mi = means[i * OUT_DIM + o];
        float ti = 0.f;
        for (int j = 0; j < 64; ++j) {
          const float s = Ao[(i << 6) + j];
          ti += s * s * means[j * OUT_DIM + o];
        }
        c0 += mi * ti;
      }
      r.x = c0; r.y = 0.f;
    }
  }
  Bf[flat] = r;
}

// ---------------------------------------------------------------------------
// GEMM: block = 8 waves tiling 32(b) x 64(o); each wave owns a 16x16 tile,
// 561 WMMAs. xs = doubled x row; xo = same rotated by one element so every
// shifted float2 LDS load is 8B-aligned for both even and odd d.
// Shifted factors for each d are preloaded into a register array so the LDS
// loads clause together and overlap the previous iteration's WMMA tail.
// ---------------------------------------------------------------------------
__global__ __launch_bounds__(256) void rbf_gemm(
    const float* __restrict__ x,   // [B][64]
    const float2* __restrict__ Bf,
    float* __restrict__ out) {     // [B][OUT]
  __shared__ __align__(16) float xs[32 * XS_STRIDE];  // xs[r][c] = x[r][c&63], c<128
  __shared__ __align__(16) float xo[32 * XS_STRIDE];  // xo[r][c] = x[r][(c+1)&63]

  const int tid    = threadIdx.x;
  const int bBlock = blockIdx.y * 32;
  const int oBlock = blockIdx.x * 64;

  for (int idx = tid; idx < 32 * 64; idx += 256) {
    const int rr = idx >> 6, cc = idx & 63;
    const float v = x[(bBlock + rr) * IN_DIM + cc];
    const int cm = (cc + 63) & 63;                    // cc-1 mod 64
    xs[rr * XS_STRIDE + cc] = v;
    xs[rr * XS_STRIDE + 64 + cc] = v;
    xo[rr * XS_STRIDE + cm] = v;
    xo[rr * XS_STRIDE + 64 + cm] = v;
  }
  __syncthreads();

  const int wave  = tid >> 5;      // 0..7
  const int lane  = tid & 31;
  const int o_sub = wave & 3;      // 0..3  -> 16 outs each
  const int b_sub = wave >> 2;     // 0..1  -> 16 b-rows each
  const int hi    = lane >> 4;
  const int row   = lane & 15;     // A-matrix row M (both lane halves)

  const float* xr  = xs + (b_sub * 16 + row) * XS_STRIDE;
  const float* xro = xo + (b_sub * 16 + row) * XS_STRIDE;
  const int ot = (blockIdx.x << 2) + o_sub;
  const float2* bp = Bf + ((long)ot * KQ_TOT) * 32 + lane;

  // d-invariant factor: 16 float2 held in registers.
  float2 xa[16];
#pragma unroll
  for (int jq = 0; jq < 16; ++jq)
    xa[jq] = *(const float2*)(xr + (jq << 2) + 2 * hi);

  v8f acc0 = {};
  v8f acc1 = {};

  int q = 0;
  for (int d = 0; d < ND; ++d) {
    // 8B-aligned base for the shifted factor (d even: xs+d ; d odd: xo+d-1).
    const float* xrB = (d & 1) ? (xro + (d - 1)) : (xr + d);

    // Preload the 16 shifted pairs -> clause of ds_load_2addr_b64, one wait.
    float2 xb[16];
#pragma unroll
    for (int jq = 0; jq < 16; ++jq)
      xb[jq] = *(const float2*)(xrB + (jq << 2) + 2 * hi);

#pragma unroll
    for (int jq = 0; jq < 16; ++jq, ++q) {
      v2f a; a.x = xa[jq].x * xb[jq].x; a.y = xa[jq].y * xb[jq].y;
      const float2 bv = bp[q * 32];                                 // 256B/wave
      v2f b; b.x = bv.x; b.y = bv.y;
      if (jq & 1)
        acc1 = __builtin_amdgcn_wmma_f32_16x16x4_f32(false, a, false, b,
                                                     (short)0, acc1, false, false);
      else
        acc0 = __builtin_amdgcn_wmma_f32_16x16x4_f32(false, a, false, b,
                                                     (short)0, acc0, false, false);
    }
  }
  // Augmented region 1: P columns = x_k, B rows = -v.
#pragma unroll
  for (int jq = 0; jq < 16; ++jq, ++q) {
    v2f a; a.x = xa[jq].x; a.y = xa[jq].y;
    const float2 bv = bp[q * 32];
    v2f b; b.x = bv.x; b.y = bv.y;
    if (jq & 1)
      acc1 = __builtin_amdgcn_wmma_f32_16x16x4_f32(false, a, false, b,
                                                   (short)0, acc1, false, false);
    else
      acc0 = __builtin_amdgcn_wmma_f32_16x16x4_f32(false, a, false, b,
                                                   (short)0, acc0, false, false);
  }
  // Augmented region 2: constant-1 column vs c0 row.
  {
    v2f a; a.x = hi ? 0.f : 1.f; a.y = 0.f;
    const float2 bv = bp[q * 32];
    v2f b; b.x = bv.x; b.y = bv.y;
    acc0 = __builtin_amdgcn_wmma_f32_16x16x4_f32(false, a, false, b,
                                                 (short)0, acc0, false, false);
  }

  // Epilogue: D layout — VGPR r holds M = r + 8*hi, N = lane&15.
  const int oIdx  = oBlock + o_sub * 16 + (lane & 15);
  const int bBase = bBlock + b_sub * 16 + 8 * hi;
#pragma unroll
  for (int r2 = 0; r2 < 8; ++r2) {
    const float qv = acc0[r2] + acc1[r2];
    out[(long)(bBase + r2) * OUT_DIM + oIdx] = __expf(-0.5f * qv);
  }
}

// ---------------------------------------------------------------------------
extern "C" void kernel_launch(void* const* d_in, const int* in_sizes, int n_in,
                              void* d_out, int out_size, void* d_ws, size_t ws_size,
                              hipStream_t stream) {
  const float* x     = (const float*)d_in[0];   // [2048][64]
  const float* means = (const float*)d_in[1];   // [64][2048]
  const float* S     = (const float*)d_in[2];   // [2048][64][64]
  float* out = (float*)d_out;                   // [2048][2048]
  float2* Bf = (float2*)d_ws;                   // 128*561*32 float2 ~= 18.4 MB

  const long total = (long)OT * KQ_TOT * 32;
  const int  threads = 256;
  const int  blocks = (int)((total + threads - 1) / threads);
  build_bfrag<<<blocks, threads, 0, stream>>>(S, means, Bf);

  dim3 grid(OUT_DIM / 64, B_DIM / 32);          // (32, 64)
  rbf_gemm<<<grid, 256, 0, stream>>>(x, Bf, out);
}